// Net_33586644255234
// MI455X (gfx1250) — compile-verified
//
#include <hip/hip_runtime.h>

typedef __attribute__((ext_vector_type(2))) float v2f;
typedef __attribute__((ext_vector_type(8))) float v8f;

#define NODES 40000
#define FIN   512
#define HID   128
#define NCLS  64

// ---------------------------------------------------------------------------
// Dual-output fp32 WMMA GEMM:  C0 = A @ B0,  C1 = A @ B1
//   A: [M x K] row-major, B0/B1: [K x Nc] row-major, C0/C1: [M x Nc].
// One wave32 computes a 16x64 slab (4 WMMA n-tiles) so the A fragment is
// reused 4x per K-step. Uses V_WMMA_F32_16X16X4_F32 (fp32 in, fp32 acc).
// M % 16 == 0, K % 4 == 0, Nc % 64 == 0 guaranteed by the problem sizes.
// ---------------------------------------------------------------------------
__global__ __launch_bounds__(256) void gemm_dual_wmma(
    const float* __restrict__ A, const float* __restrict__ B0,
    const float* __restrict__ B1, float* __restrict__ C0,
    float* __restrict__ C1, int M, int K, int Nc)
{
    const int lane = threadIdx.x & 31;
    const int wave = (blockIdx.x * blockDim.x + threadIdx.x) >> 5;
    const int groupsPerMat = Nc >> 6;           // 64-col groups per matrix
    const int groupsPerRow = groupsPerMat * 2;  // both output matrices
    const int totalGroups  = (M >> 4) * groupsPerRow;
    if (wave >= totalGroups) return;            // wave-uniform: EXEC stays all-1s

    const int mtile = wave / groupsPerRow;
    const int g     = wave % groupsPerRow;
    const float* __restrict__ B = (g < groupsPerMat) ? B0 : B1;
    float* __restrict__ C       = (g < groupsPerMat) ? C0 : C1;
    const int nbase = ((g < groupsPerMat) ? g : g - groupsPerMat) << 6;

    const int m0   = mtile << 4;
    // A 16x4 fragment layout (ISA 7.12.2): lanes 0-15 -> K=k,k+1 ; lanes 16-31 -> K=k+2,k+3
    const int arow = m0 + (lane & 15);
    const int koff = (lane >> 4) << 1;
    const int bcol = nbase + (lane & 15);

    v8f acc0 = {}, acc1 = {}, acc2 = {}, acc3 = {};
    const float* aptr = A + (long)arow * K + koff;

    for (int k = 0; k < K; k += 4) {
        v2f a;
        a.x = aptr[k];
        a.y = aptr[k + 1];

        const float* brow0 = B + (long)(k + koff) * Nc;
        const float* brow1 = brow0 + Nc;
        v2f b0, b1, b2, b3;
        b0.x = brow0[bcol];        b0.y = brow1[bcol];
        b1.x = brow0[bcol + 16];   b1.y = brow1[bcol + 16];
        b2.x = brow0[bcol + 32];   b2.y = brow1[bcol + 32];
        b3.x = brow0[bcol + 48];   b3.y = brow1[bcol + 48];

        acc0 = __builtin_amdgcn_wmma_f32_16x16x4_f32(false, a, false, b0, (short)0, acc0, false, false);
        acc1 = __builtin_amdgcn_wmma_f32_16x16x4_f32(false, a, false, b1, (short)0, acc1, false, false);
        acc2 = __builtin_amdgcn_wmma_f32_16x16x4_f32(false, a, false, b2, (short)0, acc2, false, false);
        acc3 = __builtin_amdgcn_wmma_f32_16x16x4_f32(false, a, false, b3, (short)0, acc3, false, false);
    }

    // C/D layout: VGPR v -> row m0+v (lanes 0-15) / row m0+8+v (lanes 16-31)
    const int rbase = m0 + ((lane >> 4) << 3);
    const int c0    = nbase + (lane & 15);
    for (int v = 0; v < 8; ++v) {
        long r = rbase + v;
        C[r * (long)Nc + c0]      = acc0[v];
        C[r * (long)Nc + c0 + 16] = acc1[v];
        C[r * (long)Nc + c0 + 32] = acc2[v];
        C[r * (long)Nc + c0 + 48] = acc3[v];
    }
}

// ---------------------------------------------------------------------------
// Graph pieces
// ---------------------------------------------------------------------------
__global__ void degree_kernel(const long long* __restrict__ dst,
                              float* __restrict__ deg, int E)
{
    int e = blockIdx.x * blockDim.x + threadIdx.x;
    if (e < E) atomicAdd(&deg[dst[e]], 1.0f);
}

__global__ void dinv_kernel(const float* __restrict__ deg,
                            float* __restrict__ dinv, int n)
{
    int i = blockIdx.x * blockDim.x + threadIdx.x;
    if (i < n) {
        float d = deg[i];
        dinv[i] = d > 0.0f ? rsqrtf(fmaxf(d, 1.0f)) : 0.0f;
    }
}

// One wave per edge: coalesced row gather of h[src], scaled fp32 atomic
// scatter into agg[dst]. agg (<= 20.5 MB) lives in the 192 MB L2.
__global__ __launch_bounds__(256) void scatter_kernel(
    const long long* __restrict__ src, const long long* __restrict__ dst,
    const float* __restrict__ dinv, const float* __restrict__ h,
    float* __restrict__ agg, int F, int E)
{
    int wave = (blockIdx.x * blockDim.x + threadIdx.x) >> 5;
    int lane = threadIdx.x & 31;
    if (wave >= E) return;
    long s = (long)src[wave];
    long d = (long)dst[wave];
    float norm = dinv[s] * dinv[d];
    const float* __restrict__ hs = h + s * (long)F;
    float* __restrict__ ad = agg + d * (long)F;
    for (int c = lane; c < F; c += 32)
        atomicAdd(&ad[c], norm * hs[c]);
}

__global__ void combine_relu(const float* __restrict__ agg,
                             const float* __restrict__ root,
                             const float* __restrict__ bias,
                             float* __restrict__ out, long total, int F)
{
    long i = blockIdx.x * (long)blockDim.x + threadIdx.x;
    if (i < total) {
        int c = (int)(i % F);
        out[i] = fmaxf(agg[i] + root[i] + bias[c], 0.0f);
    }
}

// ---------------------------------------------------------------------------
extern "C" void kernel_launch(void* const* d_in, const int* in_sizes, int n_in,
                              void* d_out, int out_size, void* d_ws, size_t ws_size,
                              hipStream_t stream)
{
    const float* x   = (const float*)d_in[0];
    const float* W1i = (const float*)d_in[1];
    const float* W1r = (const float*)d_in[2];
    const float* b1  = (const float*)d_in[3];
    const float* W2i = (const float*)d_in[4];
    const float* W2r = (const float*)d_in[5];
    const float* b2  = (const float*)d_in[6];
    const long long* edge = (const long long*)d_in[7];  // int64 per reference

    const int E = in_sizes[7] / 2;
    const int n = NODES;
    const long long* srcp = edge;
    const long long* dstp = edge + E;

    // Workspace carve-up (floats)
    float* ws     = (float*)d_ws;
    float* deg    = ws;                               // n
    float* dinv   = deg    + n;                       // n
    float* h_init = dinv   + n;                       // n*HID
    float* h_root = h_init + (long)n * HID;           // n*HID
    float* agg1   = h_root + (long)n * HID;           // n*HID
    float* h      = agg1   + (long)n * HID;           // n*HID
    float* g_init = h      + (long)n * HID;           // n*NCLS
    float* g_root = g_init + (long)n * NCLS;          // n*NCLS
    float* agg2   = g_root + (long)n * NCLS;          // n*NCLS

    // Zero accumulators every call (deterministic; capture-safe memset nodes)
    hipMemsetAsync(deg,  0, (size_t)n * sizeof(float), stream);
    hipMemsetAsync(agg1, 0, (size_t)n * HID * sizeof(float), stream);
    hipMemsetAsync(agg2, 0, (size_t)n * NCLS * sizeof(float), stream);

    // gcn_norm
    degree_kernel<<<(E + 255) / 256, 256, 0, stream>>>(dstp, deg, E);
    dinv_kernel<<<(n + 255) / 256, 256, 0, stream>>>(deg, dinv, n);

    // Layer 1: h_init = x@W1_init, h_root = x@W1_root   (K=512, Nc=128)
    {
        int groups = (n / 16) * ((HID >> 6) * 2);   // 2500 * 4 = 10000 waves
        int blocks = (groups + 7) / 8;              // 8 waves / 256-thr block
        gemm_dual_wmma<<<blocks, 256, 0, stream>>>(x, W1i, W1r, h_init, h_root,
                                                   n, FIN, HID);
    }
    scatter_kernel<<<(E * 32 + 255) / 256, 256, 0, stream>>>(srcp, dstp, dinv,
                                                             h_init, agg1, HID, E);
    combine_relu<<<(int)(((long)n * HID + 255) / 256), 256, 0, stream>>>(
        agg1, h_root, b1, h, (long)n * HID, HID);

    // Layer 2: g_init = h@W2_init, g_root = h@W2_root   (K=128, Nc=64)
    {
        int groups = (n / 16) * ((NCLS >> 6) * 2);  // 2500 * 2 = 5000 waves
        int blocks = (groups + 7) / 8;
        gemm_dual_wmma<<<blocks, 256, 0, stream>>>(h, W2i, W2r, g_init, g_root,
                                                   n, HID, NCLS);
    }
    scatter_kernel<<<(E * 32 + 255) / 256, 256, 0, stream>>>(srcp, dstp, dinv,
                                                             g_init, agg2, NCLS, E);
    combine_relu<<<(int)(((long)n * NCLS + 255) / 256), 256, 0, stream>>>(
        agg2, g_root, b2, (float*)d_out, (long)n * NCLS, NCLS);
}